// TemporalGeneratorNOODE_46737834115103
// MI455X (gfx1250) — compile-verified
//
#include <hip/hip_runtime.h>

// ---- problem constants (match reference) ----
#define DIM      100
#define NFRAMES  16
#define NSTEPS   15
#define BATCH    65536

// ---- tiling ----
#define NT       7            // N tiles of 16 (pad 100 -> 112)
#define KC       25           // K chunks of 4 (100 = 25*4, no pad)
#define STRIDE   116          // row stride (floats) of LDS state arrays
#define MT       2            // M tiles of 16 per wave
#define ROWS     (MT * 16)    // batch rows owned per wave (32)
#define WAVES    4            // waves per block (wave32)
#define NTHREADS (WAVES * 32)

typedef float v2f __attribute__((ext_vector_type(2)));
typedef float v8f __attribute__((ext_vector_type(8)));

// fast tanh: (e^{2x}-1)/(e^{2x}+1) with clamp; one v_exp + one rcp
__device__ __forceinline__ float fast_tanh(float x) {
  x = fminf(fmaxf(x, -15.0f), 15.0f);
  float e2 = __expf(2.0f * x);
  return (e2 - 1.0f) * __frcp_rn(e2 + 1.0f);
}

// GEMM: acc[m][jt] = A(32x100) * B(100x112); each B ds_load_b64 feeds TWO wmmas
// A layout per V_WMMA_F32_16X16X4_F32: lanes 0-15 -> (row=l, k=4kc+{0,1}),
// lanes 16-31 -> (row=l, k=4kc+{2,3}); B pre-swizzled, conflict-free ds_load_b64.
__device__ __forceinline__ void gemm_tile(const float* A_lds, const float* W_lds,
                                          int lane, v8f acc[MT][NT]) {
  const int l  = lane & 15;
  const int hi = lane >> 4;
  const float* Ab0 = A_lds + l * STRIDE + 2 * hi;
  const float* Ab1 = Ab0 + 16 * STRIDE;
#pragma unroll
  for (int m = 0; m < MT; ++m)
#pragma unroll
    for (int jt = 0; jt < NT; ++jt)
#pragma unroll
      for (int r = 0; r < 8; ++r) acc[m][jt][r] = 0.0f;

#pragma unroll 5
  for (int kc = 0; kc < KC; ++kc) {
    v2f a0 = *(const v2f*)(Ab0 + 4 * kc);
    v2f a1 = *(const v2f*)(Ab1 + 4 * kc);
#pragma unroll
    for (int jt = 0; jt < NT; ++jt) {
      v2f b = *(const v2f*)(W_lds + (jt * KC + kc) * 64 + lane * 2);
      acc[0][jt] = __builtin_amdgcn_wmma_f32_16x16x4_f32(
          false, a0, false, b, (short)0, acc[0][jt], false, false);
      acc[1][jt] = __builtin_amdgcn_wmma_f32_16x16x4_f32(
          false, a1, false, b, (short)0, acc[1][jt], false, false);
    }
  }
}

// Load W (DIM x DIM, W[j][k]) into swizzled B-operand layout, padded to 112 cols,
// zero rows k>=100. Bias padded with zeros (keeps pad columns identically 0).
__device__ __forceinline__ void load_weights(const float* __restrict__ W,
                                             const float* __restrict__ bias,
                                             float* sW, float* sB, int tid) {
  for (int i = tid; i < NT * KC * 64; i += NTHREADS) {
    int e   = i & 1;
    int ln  = (i >> 1) & 31;
    int blk = i >> 6;
    int kc  = blk % KC;
    int jt  = blk / KC;
    int k   = 4 * kc + ((ln >> 4) << 1) + e;   // B[k][j] = W[j][k]
    int j   = 16 * jt + (ln & 15);
    sW[i] = (k < DIM && j < DIM) ? W[j * DIM + k] : 0.0f;
  }
  for (int i = tid; i < NT * 16; i += NTHREADS)
    sB[i] = (i < DIM) ? bias[i] : 0.0f;
}

__global__ __launch_bounds__(NTHREADS) void noode_kernel(
    const float* __restrict__ x,
    const float* __restrict__ W0,  const float* __restrict__ b0,
    const float* __restrict__ W1,  const float* __restrict__ b1,
    const float* __restrict__ Wode, const float* __restrict__ bode,
    float* __restrict__ out) {
  __shared__ float sState[WAVES * 4 * ROWS * STRIDE];  // per wave: Y0, K1, K2, ARG
  __shared__ float sW[NT * KC * 64];                   // swizzled B operands
  __shared__ float sB[NT * 16];                        // padded bias

  const int tid  = threadIdx.x;
  const int lane = tid & 31;
  const int wave = tid >> 5;
  const int l    = lane & 15;
  const int hi   = lane >> 4;

  float* Y0  = sState + (wave * 4 + 0) * (ROWS * STRIDE);
  float* K1  = sState + (wave * 4 + 1) * (ROWS * STRIDE);
  float* K2  = sState + (wave * 4 + 2) * (ROWS * STRIDE);
  float* ARG = sState + (wave * 4 + 3) * (ROWS * STRIDE);

  const int rowBase = (blockIdx.x * WAVES + wave) * ROWS;

  // Stage this wave's x rows into ARG (pad cols 100..115 with 0).
  for (int i = lane; i < ROWS * STRIDE; i += 32) {
    int r = i / STRIDE, c = i - r * STRIDE;
    ARG[i] = (c < DIM) ? x[(size_t)(rowBase + r) * DIM + c] : 0.0f;
  }

  load_weights(W0, b0, sW, sB, tid);
  __syncthreads();

  v8f acc[MT][NT];
  v8f y1[MT][NT];

  // C/D layout: element (m, r, jt) lives at row (m*16 + r + 8*hi), col (16*jt + l)
  // base index per m-tile:
  const int cd0 = (hi * 8) * STRIDE + l;
  const int cd1 = (16 + hi * 8) * STRIDE + l;

  // ---- encoder: s0 = x @ W0^T + b0 -> Y0 (LDS) ----
  gemm_tile(ARG, sW, lane, acc);
#pragma unroll
  for (int m = 0; m < MT; ++m) {
    const int cdm = m ? cd1 : cd0;
#pragma unroll
    for (int jt = 0; jt < NT; ++jt) {
      float bb = sB[jt * 16 + l];
#pragma unroll
      for (int r = 0; r < 8; ++r)
        Y0[cdm + r * STRIDE + 16 * jt] = acc[m][jt][r] + bb;
    }
  }
  __syncthreads();

  load_weights(W1, b1, sW, sB, tid);
  __syncthreads();

  // ---- encoder: s1 = x @ W1^T + b1 -> y1 (registers) ----
  gemm_tile(ARG, sW, lane, acc);
#pragma unroll
  for (int m = 0; m < MT; ++m)
#pragma unroll
    for (int jt = 0; jt < NT; ++jt) {
      float bb = sB[jt * 16 + l];
#pragma unroll
      for (int r = 0; r < 8; ++r) y1[m][jt][r] = acc[m][jt][r] + bb;
    }

  // frame 0 = s1
#pragma unroll
  for (int m = 0; m < MT; ++m)
#pragma unroll
    for (int jt = 0; jt < NT; ++jt) {
      int j = jt * 16 + l;
      if (j < DIM) {
#pragma unroll
        for (int r = 0; r < 8; ++r) {
          int row = rowBase + m * 16 + r + 8 * hi;
          out[(size_t)row * (NFRAMES * DIM) + j] = y1[m][jt][r];
        }
      }
    }

  __syncthreads();
  load_weights(Wode, bode, sW, sB, tid);
  __syncthreads();

  const float dt  = 1.0f / (float)NSTEPS;
  const float c13 = dt * (1.0f / 3.0f);
  const float c8  = dt * 0.125f;
  const float cq  = dt * dt * 0.125f;

  for (int step = 0; step < NSTEPS; ++step) {
    // ---- k1 = tanh(Y0 W^T + b); ARG = Y0 + dt/3 * k1 ----
    gemm_tile(Y0, sW, lane, acc);
#pragma unroll
    for (int m = 0; m < MT; ++m) {
      const int cdm = m ? cd1 : cd0;
#pragma unroll
      for (int jt = 0; jt < NT; ++jt) {
        float bb = sB[jt * 16 + l];
#pragma unroll
        for (int r = 0; r < 8; ++r) {
          int idx = cdm + r * STRIDE + 16 * jt;
          float t = fast_tanh(acc[m][jt][r] + bb);
          K1[idx]  = t;
          ARG[idx] = Y0[idx] + c13 * t;
        }
      }
    }
    // ---- k2; ARG = Y0 + dt*k2 - dt/3*k1 ----
    gemm_tile(ARG, sW, lane, acc);
#pragma unroll
    for (int m = 0; m < MT; ++m) {
      const int cdm = m ? cd1 : cd0;
#pragma unroll
      for (int jt = 0; jt < NT; ++jt) {
        float bb = sB[jt * 16 + l];
#pragma unroll
        for (int r = 0; r < 8; ++r) {
          int idx = cdm + r * STRIDE + 16 * jt;
          float t   = fast_tanh(acc[m][jt][r] + bb);
          float y0v = Y0[idx];
          float k1v = K1[idx];
          K2[idx]  = t;
          ARG[idx] = y0v + dt * t - c13 * k1v;
        }
      }
    }
    // ---- k3; ARG = Y0 + dt*(k1-k2+k3); y1 += dt*Y0 + dt^2/8*(k1+2k2+k3);
    //          K1 <- k1 + 3*(k2+k3) (Q0 partial) ----
    gemm_tile(ARG, sW, lane, acc);
#pragma unroll
    for (int m = 0; m < MT; ++m) {
      const int cdm = m ? cd1 : cd0;
#pragma unroll
      for (int jt = 0; jt < NT; ++jt) {
        float bb = sB[jt * 16 + l];
#pragma unroll
        for (int r = 0; r < 8; ++r) {
          int idx = cdm + r * STRIDE + 16 * jt;
          float t   = fast_tanh(acc[m][jt][r] + bb);
          float y0v = Y0[idx];
          float k1v = K1[idx];
          float k2v = K2[idx];
          ARG[idx] = y0v + dt * (k1v - k2v + t);
          y1[m][jt][r] += dt * y0v + cq * (k1v + 2.0f * k2v + t);
          K1[idx] = k1v + 3.0f * (k2v + t);
        }
      }
    }
    // ---- k4; Y0 <- Y0 + dt/8 * (Q0 + k4) ----
    gemm_tile(ARG, sW, lane, acc);
#pragma unroll
    for (int m = 0; m < MT; ++m) {
      const int cdm = m ? cd1 : cd0;
#pragma unroll
      for (int jt = 0; jt < NT; ++jt) {
        float bb = sB[jt * 16 + l];
#pragma unroll
        for (int r = 0; r < 8; ++r) {
          int idx = cdm + r * STRIDE + 16 * jt;
          float t = fast_tanh(acc[m][jt][r] + bb);
          Y0[idx] = Y0[idx] + c8 * (K1[idx] + t);
        }
      }
    }
    // ---- frame (step+1) = y1 ----
#pragma unroll
    for (int m = 0; m < MT; ++m)
#pragma unroll
      for (int jt = 0; jt < NT; ++jt) {
        int j = jt * 16 + l;
        if (j < DIM) {
#pragma unroll
          for (int r = 0; r < 8; ++r) {
            int row = rowBase + m * 16 + r + 8 * hi;
            out[(size_t)row * (NFRAMES * DIM) + (size_t)(step + 1) * DIM + j] =
                y1[m][jt][r];
          }
        }
      }
  }
}

extern "C" void kernel_launch(void* const* d_in, const int* in_sizes, int n_in,
                              void* d_out, int out_size, void* d_ws, size_t ws_size,
                              hipStream_t stream) {
  (void)in_sizes; (void)n_in; (void)out_size; (void)d_ws; (void)ws_size;
  const float* x    = (const float*)d_in[0];
  const float* W0   = (const float*)d_in[1];
  const float* b0   = (const float*)d_in[2];
  const float* W1   = (const float*)d_in[3];
  const float* b1   = (const float*)d_in[4];
  const float* Wode = (const float*)d_in[5];
  const float* bode = (const float*)d_in[6];
  float* out = (float*)d_out;

  const int blocks = BATCH / (WAVES * ROWS);  // 65536 / 128 = 512
  noode_kernel<<<blocks, NTHREADS, 0, stream>>>(x, W0, b0, W1, b1, Wode, bode, out);
}